// GATEmbedding_83107617178432
// MI455X (gfx1250) — compile-verified
//
#include <hip/hip_runtime.h>

// ---------------------------------------------------------------------------
// GAT (3-layer) for MI455X / gfx1250.
// Memory-bound edge scatter/gather (~2 GB total -> ~90us @ 23.3 TB/s; L2-resident
// accumulators). GEMMs (~3.4 GFLOP fp32) use V_WMMA_F32_16X16X4_F32.
// ---------------------------------------------------------------------------

#define NNODES 50000
#define NEDGES 800000

typedef __attribute__((ext_vector_type(2))) float v2f;
typedef __attribute__((ext_vector_type(8))) float v8f;

__device__ __forceinline__ float leaky02(float v) { return v > 0.f ? v : 0.2f * v; }

// Order-preserving float<->int key so segment-max can use native
// global_atomic_max_i32 (monotone map; -inf -> 0x807FFFFF).
__device__ __forceinline__ int f2key(float f) {
    int i = __float_as_int(f);
    return i >= 0 ? i : (i ^ 0x7FFFFFFF);
}
__device__ __forceinline__ float key2f(int i) {
    return __int_as_float(i >= 0 ? i : (i ^ 0x7FFFFFFF));
}

// ---------------------------------------------------------------------------
// fp32 GEMM: C[M,N] = A[M,K] @ B[K,N], row-major, one wave per 16x16 tile.
// Fragment layouts per CDNA5 ISA 7.12.2:
//   A 16x4 : lane l -> M = l&15, K = 2*(l>>4) + vgpr    (2 VGPRs)
//   B 4x16 : lane l -> N = l&15, K = 2*(l>>4) + vgpr    (2 VGPRs)
//   C 16x16: lane l -> N = l&15, M = vgpr + 8*(l>>4)    (8 VGPRs)
// M (=50000) is a multiple of 16; K in {96,128} multiple of 4; N in {96,128}.
// ---------------------------------------------------------------------------
__global__ void gemm_wmma_f32(const float* __restrict__ A,
                              const float* __restrict__ B,
                              float* __restrict__ C,
                              int M, int K, int N) {
    const int wave = blockIdx.x * (blockDim.x >> 5) + (threadIdx.x >> 5);
    const int lane = threadIdx.x & 31;
    const int ntiles = N >> 4;
    const int mtiles = M >> 4;
    if (wave >= mtiles * ntiles) return;          // wave-uniform exit (EXEC all-1 at WMMA)
    const int mt = wave / ntiles;
    const int nt = wave - mt * ntiles;

    const int half  = lane >> 4;                  // 0 or 1
    const int khalf = half << 1;                  // 0 or 2
    const int mrow  = lane & 15;
    const int ncol  = lane & 15;

    const float* Arow = A + (size_t)(mt * 16 + mrow) * K;   // per-lane row
    const float* Bcol = B + nt * 16 + ncol;                 // per-lane column

    v8f acc = {};
    for (int k0 = 0; k0 < K; k0 += 4) {
        v2f a = *(const v2f*)(Arow + k0 + khalf);           // K = k0+khalf, k0+khalf+1
        v2f b;
        b.x = Bcol[(size_t)(k0 + khalf)     * N];
        b.y = Bcol[(size_t)(k0 + khalf + 1) * N];
        acc = __builtin_amdgcn_wmma_f32_16x16x4_f32(
            /*neg_a=*/false, a, /*neg_b=*/false, b,
            /*c_mod=*/(short)0, acc, /*reuse_a=*/false, /*reuse_b=*/false);
    }

    const int mbase = mt * 16 + half * 8;
    float* Cout = C + nt * 16 + ncol;
#pragma unroll
    for (int r = 0; r < 8; ++r) {
        Cout[(size_t)(mbase + r) * N] = acc[r];
    }
}

// alpha_src[n,h] = <h[n,h,:], a_s[h,:]>, alpha_dst likewise. One thread/(node,head).
__global__ void alpha_kernel(const float* __restrict__ H,
                             const float* __restrict__ a_s,
                             const float* __restrict__ a_d,
                             float* __restrict__ asrc,
                             float* __restrict__ adst,
                             int n, int heads, int c) {
    int t = blockIdx.x * blockDim.x + threadIdx.x;
    if (t >= n * heads) return;
    int node = t / heads;
    int hd   = t - node * heads;
    const float* hp  = H + (size_t)(node * heads + hd) * c;
    const float* asp = a_s + hd * c;
    const float* adp = a_d + hd * c;
    float s1 = 0.f, s2 = 0.f;
    for (int i = 0; i < c; ++i) {
        float v = hp[i];
        s1 += v * asp[i];
        s2 += v * adp[i];
    }
    asrc[t] = s1;
    adst[t] = s2;
}

__global__ void fill_f32(float* __restrict__ p, float v, int n) {
    int t = blockIdx.x * blockDim.x + threadIdx.x;
    if (t < n) p[t] = v;
}
__global__ void fill_i32(int* __restrict__ p, int v, int n) {
    int t = blockIdx.x * blockDim.x + threadIdx.x;
    if (t < n) p[t] = v;
}

// Pass A: segment max of leaky-relu logits into per-(dst,head) int keys.
__global__ void edge_max(const int* __restrict__ ei, int E, int n, int heads,
                         const float* __restrict__ asrc,
                         const float* __restrict__ adst,
                         int* __restrict__ mkey) {
    int t = blockIdx.x * blockDim.x + threadIdx.x;
    int total = (E + n) * heads;
    if (t >= total) return;
    int e  = t / heads;
    int hd = t - e * heads;
    int s, d;
    if (e < E) { s = ei[e]; d = ei[E + e]; } else { s = d = e - E; }
    float v = leaky02(asrc[s * heads + hd] + adst[d * heads + hd]);
    atomicMax(&mkey[d * heads + hd], f2key(v));
}

// Pass B: e = exp(logit - max[dst]); store per-edge; segment-sum into denom.
__global__ void edge_exp(const int* __restrict__ ei, int E, int n, int heads,
                         const float* __restrict__ asrc,
                         const float* __restrict__ adst,
                         const int* __restrict__ mkey,
                         float* __restrict__ eexp,
                         float* __restrict__ denom) {
    int t = blockIdx.x * blockDim.x + threadIdx.x;
    int total = (E + n) * heads;
    if (t >= total) return;
    int e  = t / heads;
    int hd = t - e * heads;
    int s, d;
    if (e < E) { s = ei[e]; d = ei[E + e]; } else { s = d = e - E; }
    float v  = leaky02(asrc[s * heads + hd] + adst[d * heads + hd]);
    float mx = key2f(mkey[d * heads + hd]);
    float ex = __expf(v - mx);
    eexp[t] = ex;
    atomicAdd(&denom[d * heads + hd], ex);
}

// Pass B2: normalize per-edge weights once (saves a divide per channel later).
__global__ void edge_norm(const int* __restrict__ ei, int E, int n, int heads,
                          float* __restrict__ eexp,
                          const float* __restrict__ denom) {
    int t = blockIdx.x * blockDim.x + threadIdx.x;
    int total = (E + n) * heads;
    if (t >= total) return;
    int e  = t / heads;
    int hd = t - e * heads;
    int d  = (e < E) ? ei[E + e] : (e - E);
    eexp[t] = eexp[t] / (denom[d * heads + hd] + 1e-16f);
}

// Pass C: out[dst, ch] += w[e, head(ch)] * H[src, ch]. One thread/(edge,channel);
// a wave handles 32 consecutive channels of one edge -> coalesced load + atomics.
__global__ void edge_agg(const int* __restrict__ ei, int E, int heads, int c,
                         const float* __restrict__ H,
                         const float* __restrict__ eexp,
                         float* __restrict__ out,
                         int total) {
    int t = blockIdx.x * blockDim.x + threadIdx.x;
    if (t >= total) return;
    const int F = heads * c;
    int e  = t / F;
    int ch = t - e * F;
    int hd = ch / c;
    int s, d;
    if (e < E) { s = ei[e]; d = ei[E + e]; } else { s = d = e - E; }
    float w = eexp[e * heads + hd];
    atomicAdd(&out[(size_t)d * F + ch], w * H[(size_t)s * F + ch]);
}

// out = out + bias, optional ELU (alpha=1).
__global__ void finalize_kernel(float* __restrict__ out,
                                const float* __restrict__ bias,
                                int total, int F, int do_elu) {
    int t = blockIdx.x * blockDim.x + threadIdx.x;
    if (t >= total) return;
    float v = out[t] + bias[t % F];
    if (do_elu) v = v > 0.f ? v : (__expf(v) - 1.f);
    out[t] = v;
}

// ---------------------------------------------------------------------------
// Host-side: one GAT layer.
// ---------------------------------------------------------------------------
static void run_layer(const float* X, int K,
                      const float* W, const float* a_s, const float* a_d,
                      const float* bias, int heads, int c, int do_elu,
                      const int* ei,
                      float* H, float* asrc, float* adst, int* mkey,
                      float* denom, float* eexp, float* out,
                      hipStream_t stream) {
    const int n = NNODES, E = NEDGES;
    const int F = heads * c;

    // GEMM: H = X @ W   (one wave per 16x16 tile, 4 waves per block)
    {
        int tiles = (n >> 4) * (F >> 4);
        int blocks = (tiles + 3) / 4;
        gemm_wmma_f32<<<blocks, 128, 0, stream>>>(X, W, H, n, K, F);
    }
    // attention logits per node
    {
        int t = n * heads;
        alpha_kernel<<<(t + 255) / 256, 256, 0, stream>>>(H, a_s, a_d, asrc, adst, n, heads, c);
        fill_i32<<<(t + 255) / 256, 256, 0, stream>>>(mkey, (int)0x807FFFFF, t);  // key(-inf)
        fill_f32<<<(t + 255) / 256, 256, 0, stream>>>(denom, 0.f, t);
    }
    {
        int t = n * F;
        fill_f32<<<(t + 255) / 256, 256, 0, stream>>>(out, 0.f, t);
    }
    // edge passes
    {
        int te = (E + n) * heads;
        int eb = (te + 255) / 256;
        edge_max <<<eb, 256, 0, stream>>>(ei, E, n, heads, asrc, adst, mkey);
        edge_exp <<<eb, 256, 0, stream>>>(ei, E, n, heads, asrc, adst, mkey, eexp, denom);
        edge_norm<<<eb, 256, 0, stream>>>(ei, E, n, heads, eexp, denom);
    }
    {
        int total = (E + n) * F;                       // <= 108.8M, fits int32
        edge_agg<<<(total + 255) / 256, 256, 0, stream>>>(ei, E, heads, c, H, eexp, out, total);
    }
    {
        int t = n * F;
        finalize_kernel<<<(t + 255) / 256, 256, 0, stream>>>(out, bias, t, F, do_elu);
    }
}

extern "C" void kernel_launch(void* const* d_in, const int* in_sizes, int n_in,
                              void* d_out, int out_size, void* d_ws, size_t ws_size,
                              hipStream_t stream) {
    (void)in_sizes; (void)n_in; (void)out_size; (void)ws_size;

    const float* x   = (const float*)d_in[0];
    const int*   ei  = (const int*)d_in[1];     // [2, E]: src = ei[0..E), dst = ei[E..2E)
    const float* W1  = (const float*)d_in[2];
    const float* as1 = (const float*)d_in[3];
    const float* ad1 = (const float*)d_in[4];
    const float* b1  = (const float*)d_in[5];
    const float* W2  = (const float*)d_in[6];
    const float* as2 = (const float*)d_in[7];
    const float* ad2 = (const float*)d_in[8];
    const float* b2  = (const float*)d_in[9];
    const float* W3  = (const float*)d_in[10];
    const float* as3 = (const float*)d_in[11];
    const float* ad3 = (const float*)d_in[12];
    const float* b3  = (const float*)d_in[13];

    // workspace carve-up (floats)
    float* ws   = (float*)d_ws;
    float* H    = ws;                 ws += (size_t)NNODES * 128;   // gemm output (max F=128)
    float* asrc = ws;                 ws += (size_t)NNODES * 3;
    float* adst = ws;                 ws += (size_t)NNODES * 3;
    int*   mkey = (int*)ws;           ws += (size_t)NNODES * 3;
    float* den  = ws;                 ws += (size_t)NNODES * 3;
    float* eexp = ws;                 ws += (size_t)(NEDGES + NNODES) * 3;
    float* agg1 = ws;                 ws += (size_t)NNODES * 96;
    float* agg2 = ws;                 ws += (size_t)NNODES * 96;

    // layer 1: in=128 -> 3 heads x 32, concat, ELU
    run_layer(x,    128, W1, as1, ad1, b1, 3, 32, /*elu=*/1, ei, H, asrc, adst, mkey, den, eexp, agg1, stream);
    // layer 2: in=96 -> 3 heads x 32, concat, ELU
    run_layer(agg1,  96, W2, as2, ad2, b2, 3, 32, /*elu=*/1, ei, H, asrc, adst, mkey, den, eexp, agg2, stream);
    // layer 3: in=96 -> 1 head x 128, mean over 1 head == identity, no ELU -> d_out
    run_layer(agg2,  96, W3, as3, ad3, b3, 1, 128, /*elu=*/0, ei, H, asrc, adst, mkey, den, eexp, (float*)d_out, stream);
}